// MRConv3d_5016521801766
// MI455X (gfx1250) — compile-verified
//
#include <hip/hip_runtime.h>
#include <hip/hip_bf16.h>
#include <math.h>
#include <stdint.h>

// ---- problem constants (match reference) ----
#define BB    2
#define CC    192
#define DD    16
#define HH    32
#define WW    32
#define NN    (DD * HH * WW)     // 16384
#define KK    9
#define OC    384                // out channels
#define TWO_C (2 * CC)           // 384 (GEMM K-dim)
#define KT    (TWO_C / 4)        // 96 K-steps of 4

typedef float v2f __attribute__((ext_vector_type(2)));
typedef float v4f __attribute__((ext_vector_type(4)));
typedef float v8f __attribute__((ext_vector_type(8)));

// -----------------------------------------------------------------------------
// Kernel 0: tiled transpose x[b][c][n] -> xT[b][n][c] so the per-node gathers
// in the aggregation become lane-contiguous 128B requests instead of 32 random
// cachelines per instruction (~16x L2 bandwidth saving on the gather stage).
// -----------------------------------------------------------------------------
__global__ void __launch_bounds__(256)
transpose_x(const float* __restrict__ x, float* __restrict__ xT) {
  __shared__ float tile[32][33];
  const int b  = blockIdx.z;
  const int c0 = blockIdx.y * 32;
  const int n0 = blockIdx.x * 32;
  const int tx = threadIdx.x;            // 0..31
  const int ty = threadIdx.y;            // 0..7
  const float* xb = x + (long)b * CC * NN;
#pragma unroll
  for (int r = 0; r < 4; ++r) {
    const int c = ty + r * 8;
    tile[c][tx] = xb[(long)(c0 + c) * NN + n0 + tx];   // coalesced over n
  }
  __syncthreads();
  float* xo = xT + (long)b * NN * CC;
#pragma unroll
  for (int r = 0; r < 4; ++r) {
    const int n = ty + r * 8;
    xo[(long)(n0 + n) * CC + c0 + tx] = tile[tx][n];   // coalesced over c
  }
}

// -----------------------------------------------------------------------------
// Kernel 1: max-relative aggregation (node-major gathers, channel-major output).
// Phase A: lanes = channels -> every gather xT[idx*C + c0+lane] is one
//          contiguous 128B wave request; per-node results parked in LDS.
// Phase B: lanes = nodes -> paired featP[b][c][n] = (x, max_k(xj-xi)) written
//          with coalesced 256B stores.
// -----------------------------------------------------------------------------
__global__ void __launch_bounds__(256)
mr_aggregate(const float* __restrict__ xT, const int* __restrict__ ei,
             float* __restrict__ featP) {
  __shared__ int sj[32 * KK];
  __shared__ int si[32 * KK];
  __shared__ v2f lp[CC * 33];            // [c][node], stride 33 kills bank conflicts

  const int b  = blockIdx.y;
  const int n0 = blockIdx.x * 32;
  const int t  = threadIdx.x;

  // edge_index layout [2][B][N][K]; e=0 -> neighbor (x_j), e=1 -> center (x_i)
  const long ebase_j = ((long)b * NN + n0) * KK;
  const long ebase_i = ((long)(BB + b) * NN + n0) * KK;
  for (int u = t; u < 32 * KK; u += 256) {
    sj[u] = ei[ebase_j + u];
    si[u] = ei[ebase_i + u];
  }
  __syncthreads();

  const int lane = t & 31;               // channel within chunk
  const int w    = t >> 5;               // wave id: owns 4 nodes
  const float* xb = xT + (long)b * NN * CC;

  for (int q = 0; q < 4; ++q) {
    const int nl = w * 4 + q;
    const float* xcen = xb + (long)(n0 + nl) * CC;
    for (int cc = 0; cc < CC / 32; ++cc) {
      const int c = cc * 32 + lane;
      float m = -INFINITY;
#pragma unroll
      for (int k = 0; k < KK; ++k) {
        const float xjv = xb[(long)sj[nl * KK + k] * CC + c];  // 128B coalesced
        const float xiv = xb[(long)si[nl * KK + k] * CC + c];
        m = fmaxf(m, xjv - xiv);
      }
      v2f pr;
      pr.x = xcen[c];                    // GEMM K-row 2c
      pr.y = m;                          // GEMM K-row 2c+1
      lp[c * 33 + nl] = pr;
    }
  }
  __syncthreads();

  // Phase B: coalesced write-out, lanes over nodes
  v2f* fb = (v2f*)featP + (long)b * CC * NN;
  const int nlB = t & 31;
  for (int c = t >> 5; c < CC; c += 8) {
    fb[(long)c * NN + n0 + nlB] = lp[c * 33 + nlB];
  }
}

// -----------------------------------------------------------------------------
// Kernel 2: repack weight [OC][2C] into M=32 WMMA A-panel layout.
// For row-block mt32 (32 rows): lane l, sub-tile s(=rows +16*s), vgpr v holds
//   w[mt32*32 + s*16 + (l&15)][kt*4 + v + 2*(l>>4)]
// stored as [mt32][kt][lane][s] float2 -> a wave fetches BOTH A sub-tiles for a
// k-step with one 16B LDS read after the panel is async-staged.
// -----------------------------------------------------------------------------
__global__ void __launch_bounds__(256)
repack_weight(const float* __restrict__ w, float* __restrict__ wrep) {
  const int gid = blockIdx.x * blockDim.x + threadIdx.x;
  const int total = (OC / 32) * KT * 32 * 2;             // 73728
  if (gid >= total) return;
  const int s    = gid & 1;
  const int lane = (gid >> 1) & 31;
  const int kt   = (gid >> 6) % KT;
  const int mt32 = (gid >> 6) / KT;
  const int m    = mt32 * 32 + s * 16 + (lane & 15);
  const int kb   = kt * 4 + 2 * (lane >> 4);
  wrep[(size_t)gid * 2 + 0] = w[(long)m * TWO_C + kb + 0];
  wrep[(size_t)gid * 2 + 1] = w[(long)m * TWO_C + kb + 1];
}

// -----------------------------------------------------------------------------
// Kernel 3: GEMM  out[b][o][n] = relu( sum_k w[o][k]*feat[k][n] + bias[o] )
// via V_WMMA_F32_16X16X4_F32, M=32 x N=64 per wave:
//  - 48KB A panel (shared by all 8 waves, same mt32) async-staged to LDS once
//    per block (global_load_async_to_lds_b128 + s_wait_asynccnt).
//  - k-loop: 1x ds_load_b128 (both A sub-tiles) + 4x global b64 (B K-pairs,
//    paired feat layout) feeding 8 WMMAs -> 2 WMMAs per B-load.
// -----------------------------------------------------------------------------
__global__ void __launch_bounds__(256)
mr_gemm(const float* __restrict__ wrep, const float* __restrict__ featP,
        const float* __restrict__ bias, float* __restrict__ out) {
  __shared__ v2f sA[KT * 32 * 2];                       // 49152 bytes

  const int t    = threadIdx.x;
  const int lane = t & 31;
  const int wave = t >> 5;                              // 0..7
  const int b    = blockIdx.z;
  const int mt32 = blockIdx.y;                          // 0..11
  const int n0   = (blockIdx.x * 8 + wave) * 64;

  // ---- async-stage this mt32's A panel into LDS ----
  {
    const char* gsrc = (const char*)(wrep + (size_t)mt32 * KT * 32 * 4); // 49152 B
#pragma unroll
    for (int it = 0; it < 12; ++it) {                   // 12*256*16B = 49152 B
      const uint32_t off = (uint32_t)(it * 256 + t) * 16u;
      const uint32_t ldsoff = (uint32_t)(uintptr_t)(&sA[0]) + off;
      const uint64_t ga = (uint64_t)(uintptr_t)(gsrc + off);
      asm volatile("global_load_async_to_lds_b128 %0, %1, off"
                   :: "v"(ldsoff), "v"(ga) : "memory");
    }
    asm volatile("s_wait_asynccnt 0x0" ::: "memory");
  }
  __syncthreads();

  v8f acc0[4], acc1[4];
#pragma unroll
  for (int j = 0; j < 4; ++j) { acc0[j] = (v8f){}; acc1[j] = (v8f){}; }

  // B operand: lane needs K-pair p = kt*2 + (lane>>4) at column n0+(lane&15).
  const v2f* fp = (const v2f*)featP + (long)b * CC * NN
                + (long)(lane >> 4) * NN + (n0 + (lane & 15));

  for (int kt = 0; kt < KT; ++kt) {
    const v4f av = ((const v4f*)sA)[kt * 32 + lane];    // ds_load_b128: A s=0,1
    v2f a0, a1;
    a0.x = av.x; a0.y = av.y;
    a1.x = av.z; a1.y = av.w;
    const v2f* f0 = fp + (long)(kt * 2) * NN;
    __builtin_prefetch(f0 + 2 * NN, 0, 3);              // next K-step's B stream
#pragma unroll
    for (int j = 0; j < 4; ++j) {
      const v2f bv = f0[j * 16];                        // one b64 feeds 2 WMMAs
      acc0[j] = __builtin_amdgcn_wmma_f32_16x16x4_f32(
          false, a0, false, bv, (short)0, acc0[j], false, false);
      acc1[j] = __builtin_amdgcn_wmma_f32_16x16x4_f32(
          false, a1, false, bv, (short)0, acc1[j], false, false);
    }
  }

  // Epilogue: VGPR r holds M = base + s*16 + r + 8*(lane>>4), N = lane&15.
  float* ob = out + (long)b * OC * NN;
  const int col = n0 + (lane & 15);
#pragma unroll
  for (int s = 0; s < 2; ++s) {
    const int m0 = mt32 * 32 + s * 16 + (lane >> 4) * 8;
#pragma unroll
    for (int r = 0; r < 8; ++r) {
      const int   row = m0 + r;
      const float bs  = bias[row];
#pragma unroll
      for (int j = 0; j < 4; ++j) {
        float v = (s ? acc1[j][r] : acc0[j][r]) + bs;
        v = v > 0.0f ? v : 0.0f;                        // ReLU
        ob[(long)row * NN + col + j * 16] = v;
      }
    }
  }
}

// -----------------------------------------------------------------------------
// Launch
// -----------------------------------------------------------------------------
extern "C" void kernel_launch(void* const* d_in, const int* in_sizes, int n_in,
                              void* d_out, int out_size, void* d_ws, size_t ws_size,
                              hipStream_t stream) {
  const float* x    = (const float*)d_in[0];   // [B,C,D,H,W]
  const int*   ei   = (const int*)d_in[1];     // [2,B,N,K]
  const float* wgt  = (const float*)d_in[2];   // [OUT, 2C]
  const float* bias = (const float*)d_in[3];   // [OUT]
  float*       out  = (float*)d_out;           // [B,OUT,D,H,W]

  // workspace: xT [B][N][C], paired feat [B][C][N][2], repacked weight panels
  float* xT    = (float*)d_ws;                          // B*N*C floats
  float* featP = xT + (size_t)BB * NN * CC;             // B*C*N*2 floats
  float* wrep  = featP + (size_t)BB * CC * NN * 2;      // OC/32*KT*128 floats

  // 0) transpose x -> node-major
  {
    dim3 grid(NN / 32, CC / 32, BB);
    transpose_x<<<grid, dim3(32, 8), 0, stream>>>(x, xT);
  }
  // 1) aggregation -> paired feat
  {
    dim3 grid(NN / 32, BB);
    mr_aggregate<<<grid, 256, 0, stream>>>(xT, ei, featP);
  }
  // 2) weight repack into M=32 WMMA A-panel layout
  {
    const int total = (OC / 32) * KT * 32 * 2;          // 73728
    repack_weight<<<(total + 255) / 256, 256, 0, stream>>>(wgt, wrep);
  }
  // 3) WMMA GEMM + bias + ReLU
  {
    dim3 grid(NN / (64 * 8), OC / 32, BB);              // (32, 12, 2)
    mr_gemm<<<grid, 256, 0, stream>>>(wrep, featP, bias, out);
  }
}